// HybridQuantumNet_39367670235140
// MI455X (gfx1250) — compile-verified
//
#include <hip/hip_runtime.h>
#include <math.h>

typedef __attribute__((ext_vector_type(2))) float v2f;
typedef __attribute__((ext_vector_type(8))) float v8f;

#define NQ 4
#define DIM 16
#define NLAYERS 2
#define BLOCK 256
#define WAVES (BLOCK / 32)

struct cplx { float re, im; };
__device__ inline cplx cmul(cplx a, cplx b) { return {a.re*b.re - a.im*b.im, a.re*b.im + a.im*b.re}; }
__device__ inline cplx cadd(cplx a, cplx b) { return {a.re + b.re, a.im + b.im}; }

// ---------------------------------------------------------------------------
// Setup: build V = U * diag((-i)^popcount(a)) from the circuit weights.
// Thread a (0..15) propagates basis column e_a through Rot gates + CNOT rings.
// Bit convention: wire w <-> bit (3-w), i.e. wire0 is the MSB of the 4-bit
// amplitude index (matches jnp state [B,2,2,2,2] with wire w at axis 1+w).
// ---------------------------------------------------------------------------
__global__ void build_V_kernel(const float* __restrict__ weights,
                               float* __restrict__ Vr, float* __restrict__ Vi)
{
    int a = threadIdx.x;
    if (a >= DIM) return;

    cplx s[DIM];
    for (int b = 0; b < DIM; ++b) s[b] = { (b == a) ? 1.f : 0.f, 0.f };

    for (int l = 0; l < NLAYERS; ++l) {
        for (int q = 0; q < NQ; ++q) {
            float phi   = weights[(l*NQ + q)*3 + 0];
            float theta = weights[(l*NQ + q)*3 + 1];
            float omega = weights[(l*NQ + q)*3 + 2];
            float ct = cosf(0.5f*theta), st = sinf(0.5f*theta);
            // Rot = RZ(omega) RY(theta) RZ(phi):
            // g00 = ct e^{-iA}, g01 = -st e^{-iB}, g10 = st e^{iB}, g11 = ct e^{iA}
            // with A = (phi+omega)/2, B = (omega-phi)/2
            float A = 0.5f*(phi + omega), Bv = 0.5f*(omega - phi);
            float cA = cosf(A), sA = sinf(A), cB = cosf(Bv), sB = sinf(Bv);
            cplx g00 = { ct*cA, -ct*sA };
            cplx g01 = { -st*cB,  st*sB };
            cplx g10 = {  st*cB,  st*sB };
            cplx g11 = {  ct*cA,  ct*sA };
            int mask = 8 >> q;
            for (int b0 = 0; b0 < DIM; ++b0) {
                if (b0 & mask) continue;
                int b1 = b0 | mask;
                cplx v0 = s[b0], v1 = s[b1];
                s[b0] = cadd(cmul(g00, v0), cmul(g01, v1));
                s[b1] = cadd(cmul(g10, v0), cmul(g11, v1));
            }
        }
        int r = l % (NQ - 1) + 1;   // layer 0: r=1, layer 1: r=2
        for (int q = 0; q < NQ; ++q) {
            int cm = 8 >> q;
            int tm = 8 >> ((q + r) % NQ);
            for (int b = 0; b < DIM; ++b) {
                if ((b & cm) && !(b & tm)) {
                    cplx t = s[b]; s[b] = s[b | tm]; s[b | tm] = t;
                }
            }
        }
    }

    // fold phase of column a: (-i)^popcount(a)
    int p = __popc(a) & 3;
    cplx ph;
    switch (p) { case 0: ph = {1.f,0.f}; break; case 1: ph = {0.f,-1.f}; break;
                 case 2: ph = {-1.f,0.f}; break; default: ph = {0.f,1.f}; }
    for (int b = 0; b < DIM; ++b) {
        cplx v = cmul(s[b], ph);
        Vr[b*DIM + a] = v.re;
        Vi[b*DIM + a] = v.im;
    }
}

// ---------------------------------------------------------------------------
// Main kernel: per wave, 32 batch elements -> two 16x16 A tiles, real+imag
// GEMMs on the matrix pipe via chained V_WMMA_F32_16X16X4_F32, then |psi|^2
// and the sign-weighted PauliZ/Linear reduction.
// ---------------------------------------------------------------------------
__global__ void __launch_bounds__(BLOCK) qcirc_kernel(
    const float* __restrict__ x,
    const float* __restrict__ Vr,
    const float* __restrict__ Vi,
    const float* __restrict__ fc_w,
    const float* __restrict__ fc_b,
    float* __restrict__ out,
    int B)
{
    __shared__ float Rbuf[WAVES][32][16];   // per-wave batch-row magnitudes r
    __shared__ float Pbuf[WAVES][32][16];   // per-wave probabilities

    const int lane = threadIdx.x & 31;
    const int wave = threadIdx.x >> 5;
    const int e    = blockIdx.x * BLOCK + threadIdx.x;
    const int ec   = (e < B) ? e : (B - 1);   // clamp: keep EXEC all-ones for WMMA

    // --- product-state magnitudes r[16] for this element ---
    float4 xv = reinterpret_cast<const float4*>(x)[ec];
    float s0, c0, s1, c1, s2, c2, s3, c3;
    __sincosf(0.5f*xv.x, &s0, &c0);
    __sincosf(0.5f*xv.y, &s1, &c1);
    __sincosf(0.5f*xv.z, &s2, &c2);
    __sincosf(0.5f*xv.w, &s3, &c3);
    float q01[4] = { c0*c1, c0*s1, s0*c1, s0*s1 };   // (wire0,wire1) bits
    float q23[4] = { c2*c3, c2*s3, s2*c3, s2*s3 };   // (wire2,wire3) bits
    float* rrow = Rbuf[wave][lane];
    #pragma unroll
    for (int i = 0; i < 4; ++i) {
        float4 v = make_float4(q01[i]*q23[0], q01[i]*q23[1],
                               q01[i]*q23[2], q01[i]*q23[3]);
        reinterpret_cast<float4*>(rrow)[i] = v;     // r[a], a = i<<2 | j
    }
    __syncthreads();

    const int n  = lane & 15;
    const int hi = lane >> 4;

    // B fragments (constant, L2-resident): W = Vr^T, B_t[k][n] = Vr[n][4t+k]
    // 4x16 f32 B layout assumed to mirror A: N = lane&15, VGPR j = K (2*hi..)
    v2f bre[4], bim[4];
    #pragma unroll
    for (int t = 0; t < 4; ++t) {
        int k0 = 4*t + 2*hi;
        bre[t] = *reinterpret_cast<const v2f*>(Vr + n*16 + k0);
        bim[t] = *reinterpret_cast<const v2f*>(Vi + n*16 + k0);
    }

    float w00 = fc_w[0], w01 = fc_w[1], w10 = fc_w[2], w11 = fc_w[3];

    #pragma unroll
    for (int tile = 0; tile < 2; ++tile) {
        // A fragments (16x4 f32): lanes 0-15 hold M=lane, K=4t+{0,1};
        // lanes 16-31 hold K=4t+{2,3}
        v2f afrag[4];
        const float* abase = &Rbuf[wave][tile*16 + n][0];
        #pragma unroll
        for (int t = 0; t < 4; ++t)
            afrag[t] = *reinterpret_cast<const v2f*>(abase + 4*t + 2*hi);

        v8f accR = {0,0,0,0,0,0,0,0};
        v8f accI = {0,0,0,0,0,0,0,0};
        #pragma unroll
        for (int t = 0; t < 4; ++t) {
            accR = __builtin_amdgcn_wmma_f32_16x16x4_f32(
                       false, afrag[t], false, bre[t], (short)0, accR, false, false);
            accI = __builtin_amdgcn_wmma_f32_16x16x4_f32(
                       false, afrag[t], false, bim[t], (short)0, accI, false, false);
        }
        // prob = |psi|^2 ; scatter into per-element rows via C/D layout:
        // D[m][n]: n = lane&15, m = v + 8*(lane>=16)
        #pragma unroll
        for (int v = 0; v < 8; ++v) {
            float pr = accR[v]*accR[v] + accI[v]*accI[v];
            Pbuf[wave][tile*16 + v + 8*hi][n] = pr;
        }
    }
    __syncthreads();

    // z0 = sum prob * sign(bit3), z1 = sum prob * sign(bit2); fold Linear(2,2)
    float o0 = fc_b[0], o1 = fc_b[1];
    const float* prow = Pbuf[wave][lane];
    #pragma unroll
    for (int b = 0; b < 16; ++b) {
        float sz0 = (b & 8) ? -1.f : 1.f;   // wire 0 (MSB)
        float sz1 = (b & 4) ? -1.f : 1.f;   // wire 1
        float p = prow[b];
        o0 = fmaf(p, w00*sz0 + w01*sz1, o0);
        o1 = fmaf(p, w10*sz0 + w11*sz1, o1);
    }
    if (e < B)
        reinterpret_cast<float2*>(out)[e] = make_float2(o0, o1);
}

extern "C" void kernel_launch(void* const* d_in, const int* in_sizes, int n_in,
                              void* d_out, int out_size, void* d_ws, size_t ws_size,
                              hipStream_t stream)
{
    const float* x       = (const float*)d_in[0];   // [B,4]
    const float* weights = (const float*)d_in[1];   // [2,4,3]
    const float* fc_w    = (const float*)d_in[2];   // [2,2]
    const float* fc_b    = (const float*)d_in[3];   // [2]
    float* out = (float*)d_out;                     // [B,2]
    int B = in_sizes[0] / 4;

    float* Vr = (float*)d_ws;        // 16x16 f32
    float* Vi = Vr + DIM*DIM;        // 16x16 f32

    build_V_kernel<<<1, 32, 0, stream>>>(weights, Vr, Vi);
    int grid = (B + BLOCK - 1) / BLOCK;
    qcirc_kernel<<<grid, BLOCK, 0, stream>>>(x, Vr, Vi, fc_w, fc_b, out, B);
}